// MFPVMLayer_72344429134361
// MI455X (gfx1250) — compile-verified
//
#include <hip/hip_runtime.h>
#include <math.h>

// ---------------- problem constants ----------------
#define BR    4          // branches
#define NB    8          // batch
#define LSEQ  1024       // 32*32 spatial -> sequence length
#define CM    64         // D_MODEL
#define DI    128        // D_INNER
#define DS    16         // D_STATE
#define DCONV 4
#define DTR   4
#define NXP   36         // DT_RANK + 2*D_STATE
#define EPSF  1e-5f
#define TOKB  (NB*LSEQ)  // tokens per branch = 8192

typedef __attribute__((ext_vector_type(2))) float v2f;
typedef __attribute__((ext_vector_type(8))) float v8f;
typedef __attribute__((ext_vector_type(4))) int   v4i;

// ---------------- workspace layout (float offsets) ----------------
#define OFF_XN   ((size_t)0)
#define OFF_XZ   (OFF_XN   + (size_t)BR*TOKB*CM)
#define OFF_XC   (OFF_XZ   + (size_t)BR*TOKB*2*DI)
#define OFF_PROJ (OFF_XC   + (size_t)BR*TOKB*DI)
#define OFF_DT   (OFF_PROJ + (size_t)BR*TOKB*NXP)
#define OFF_YZ   (OFF_DT   + (size_t)BR*TOKB*DI)
// total ~24.25M floats (~97 MB)

// ---------------- CDNA5 async global->LDS copy (guarded) ----------------
#if defined(__has_builtin)
#if __has_builtin(__builtin_amdgcn_global_load_async_to_lds_b128) && \
    __has_builtin(__builtin_amdgcn_s_wait_asynccnt)
#define USE_ASYNC_LDS 1
#endif
#endif
#ifndef USE_ASYNC_LDS
#define USE_ASYNC_LDS 0
#endif

__device__ __forceinline__ void cp_b128(float* lds_dst, const float* gsrc) {
#if USE_ASYNC_LDS
    // builtin signature (per hipcc diagnostic): (v4i AS1*, v4i AS3*, imm offset, imm cpol)
    __builtin_amdgcn_global_load_async_to_lds_b128(
        (__attribute__((address_space(1))) v4i*)gsrc,
        (__attribute__((address_space(3))) v4i*)lds_dst, 0, 0);
#else
    *(float4*)lds_dst = *(const float4*)gsrc;
#endif
}

// =================================================================
// 1) LayerNorm over channels; (B,C,H,W) -> token-major xn[br][b][l][c]
// =================================================================
__global__ __launch_bounds__(256) void ln_kernel(
    const float* __restrict__ x0, const float* __restrict__ x1,
    const float* __restrict__ x2, const float* __restrict__ x3,
    const float* __restrict__ g0, const float* __restrict__ g1,
    const float* __restrict__ g2, const float* __restrict__ g3,
    const float* __restrict__ be0, const float* __restrict__ be1,
    const float* __restrict__ be2, const float* __restrict__ be3,
    float* __restrict__ ws)
{
    __shared__ float tile[CM][65];
    __shared__ float s_mean[64], s_rstd[64];
    int blk = blockIdx.x;              // br*128 + b*16 + lt
    int br = blk >> 7;
    int b  = (blk >> 4) & 7;
    int lt = blk & 15;
    int l0 = lt * 64;
    const float* xin = (br==0)?x0:(br==1)?x1:(br==2)?x2:x3;
    const float* g   = (br==0)?g0:(br==1)?g1:(br==2)?g2:g3;
    const float* be  = (br==0)?be0:(br==1)?be1:(br==2)?be2:be3;
    int t = threadIdx.x;
    int li = t & 63, cbase = t >> 6;
    const float* src = xin + (size_t)b*CM*LSEQ + l0;
    for (int c = cbase; c < CM; c += 4)                 // coalesced along l
        tile[c][li] = src[(size_t)c*LSEQ + li];
    __syncthreads();
    if (t < 64) {
        float s = 0.f, s2 = 0.f;
        #pragma unroll 8
        for (int c = 0; c < CM; ++c) { float v = tile[c][t]; s += v; s2 += v*v; }
        float m = s * (1.f/CM);
        float var = s2 * (1.f/CM) - m*m;
        s_mean[t] = m;
        s_rstd[t] = rsqrtf(var + EPSF);
    }
    __syncthreads();
    float* xn = ws + OFF_XN + ((size_t)(br*NB + b)*LSEQ + l0)*CM;
    for (int idx = t; idx < 64*CM; idx += 256) {
        int l = idx >> 6, c = idx & 63;
        xn[(size_t)l*CM + c] = (tile[c][l] - s_mean[l]) * s_rstd[l] * g[c] + be[c];
    }
}

// =================================================================
// 2) xz = xn @ W_in^T  (M=8192/branch, N=256, K=64) via f32 WMMA
// =================================================================
__global__ __launch_bounds__(256) void gemm_in_kernel(
    const float* __restrict__ W_in, float* __restrict__ ws)
{
    __shared__ float aL[16][68];                        // bank-conflict-free
    int blk = blockIdx.x;                               // br*512 + mtile
    int br = blk >> 9, mt = blk & 511;
    size_t tokbase = (size_t)br*TOKB + (size_t)mt*16;
    const float* xn = ws + OFF_XN + tokbase*CM;
    int t = threadIdx.x;
    for (int idx = t; idx < 16*CM; idx += 256)
        aL[idx >> 6][idx & 63] = xn[idx];
    __syncthreads();
    int wave = t >> 5, lane = t & 31;
    int m = lane & 15, kh = lane >> 4, n = lane & 15;
    float* xz = ws + OFF_XZ + tokbase*(2*DI);
    for (int nt = wave; nt < 16; nt += 8) {             // each wave: 2 N-tiles
        int n0 = nt * 16;
        v8f acc = {};
        #pragma unroll
        for (int k0 = 0; k0 < CM; k0 += 4) {
            int k = k0 + kh*2;
            v2f a, bb;
            a.x = aL[m][k]; a.y = aL[m][k+1];
            bb.x = W_in[(size_t)(n0+n)*CM + k];
            bb.y = W_in[(size_t)(n0+n)*CM + k + 1];
            acc = __builtin_amdgcn_wmma_f32_16x16x4_f32(
                false, a, false, bb, (short)0, acc, false, false);
        }
        #pragma unroll
        for (int r = 0; r < 8; ++r)
            xz[(size_t)(r + 8*kh)*(2*DI) + n0 + n] = acc[r];
    }
}

// =================================================================
// 3) causal depthwise conv(4) + SiLU -> xc
// =================================================================
__global__ __launch_bounds__(256) void conv_silu_kernel(
    const float* __restrict__ W_conv, const float* __restrict__ b_conv,
    float* __restrict__ ws)
{
    size_t idx = (size_t)blockIdx.x*256 + threadIdx.x;  // over br*b*l*d
    int d = (int)(idx & (DI-1));
    size_t tl = idx >> 7;                               // global token
    int l = (int)(tl & (LSEQ-1));
    const float* xi = ws + OFF_XZ;                      // stride 256, cols 0..127
    float acc = b_conv[d];
    #pragma unroll
    for (int k = 0; k < DCONV; ++k) {
        int ls = l + k - (DCONV-1);
        float v = (ls >= 0) ? xi[((size_t)((long)tl + k - (DCONV-1)))*(2*DI) + d] : 0.f;
        acc = fmaf(v, W_conv[d*DCONV + k], acc);
    }
    ws[OFF_XC + tl*DI + d] = acc / (1.f + expf(-acc));  // silu
}

// =================================================================
// 4a) proj = xc @ W_xproj^T (N=36 -> 3 WMMA N-tiles, masked cols)
// =================================================================
__global__ __launch_bounds__(128) void gemm_proj_kernel(
    const float* __restrict__ W_xproj, float* __restrict__ ws)
{
    __shared__ float aL[16][132];
    int blk = blockIdx.x;
    int br = blk >> 9, mt = blk & 511;
    size_t tokbase = (size_t)br*TOKB + (size_t)mt*16;
    const float* xc = ws + OFF_XC + tokbase*DI;
    int t = threadIdx.x;
    for (int idx = t; idx < 16*DI; idx += 128)
        aL[idx >> 7][idx & 127] = xc[idx];
    __syncthreads();
    int wave = t >> 5, lane = t & 31;
    if (wave < 3) {                                     // wave-uniform branch
        int m = lane & 15, kh = lane >> 4, n = lane & 15;
        int nn = wave*16 + n;
        float bmask = (nn < NXP) ? 1.f : 0.f;           // mask, no divergence
        int nc = (nn < NXP) ? nn : 0;
        v8f acc = {};
        #pragma unroll
        for (int k0 = 0; k0 < DI; k0 += 4) {
            int k = k0 + kh*2;
            v2f a, bb;
            a.x = aL[m][k]; a.y = aL[m][k+1];
            bb.x = W_xproj[(size_t)nc*DI + k]     * bmask;
            bb.y = W_xproj[(size_t)nc*DI + k + 1] * bmask;
            acc = __builtin_amdgcn_wmma_f32_16x16x4_f32(
                false, a, false, bb, (short)0, acc, false, false);
        }
        float* proj = ws + OFF_PROJ + tokbase*NXP;
        if (nn < NXP)
            for (int r = 0; r < 8; ++r)
                proj[(size_t)(r + 8*kh)*NXP + nn] = acc[r];
    }
}

// =================================================================
// 4b) dt = softplus(proj[:, :4] @ W_dt^T + b_dt) — single K=4 WMMA
// =================================================================
__global__ __launch_bounds__(256) void dt_kernel(
    const float* __restrict__ W_dt, const float* __restrict__ b_dt,
    float* __restrict__ ws)
{
    int blk = blockIdx.x;
    int br = blk >> 9, mt = blk & 511;
    size_t tokbase = (size_t)br*TOKB + (size_t)mt*16;
    const float* proj = ws + OFF_PROJ + tokbase*NXP;
    int t = threadIdx.x;
    int wave = t >> 5, lane = t & 31;
    int m = lane & 15, kh = lane >> 4, n = lane & 15;
    int n0 = wave * 16;                                 // 8 waves cover N=128
    v2f a, bb;
    a.x = proj[(size_t)m*NXP + kh*2];
    a.y = proj[(size_t)m*NXP + kh*2 + 1];
    bb.x = W_dt[(n0+n)*DTR + kh*2];
    bb.y = W_dt[(n0+n)*DTR + kh*2 + 1];
    v8f acc = {};
    acc = __builtin_amdgcn_wmma_f32_16x16x4_f32(
        false, a, false, bb, (short)0, acc, false, false);
    float* dt = ws + OFF_DT + tokbase*DI;
    #pragma unroll
    for (int r = 0; r < 8; ++r) {
        float v = acc[r] + b_dt[n0 + n];
        float sp = (v > 20.f) ? v : log1pf(expf(v));    // stable softplus
        dt[(size_t)(r + 8*kh)*DI + n0 + n] = sp;
    }
}

// =================================================================
// 5) selective scan, restructured:
//    - per-iteration critical path = v_exp + v_fma only
//    - h·C parked in wave-private LDS, reduced in bulk per 32-step chunk
//    - dt/xc/B/C double-buffered in LDS via async global->LDS b128 copies
//      (ASYNCcnt threshold wait overlaps next-chunk fill with compute)
// =================================================================
#define TCH 32
#define NCH (LSEQ/TCH)

__global__ __launch_bounds__(256) void scan_kernel(
    const float* __restrict__ A_log, const float* __restrict__ Dp,
    float* __restrict__ ws)
{
    __shared__ __align__(16) float sdt[2][TCH][16];
    __shared__ __align__(16) float sxc[2][TCH][16];
    __shared__ __align__(16) float sB [2][TCH][16];
    __shared__ __align__(16) float sC [2][TCH][16];
    __shared__ __align__(16) float sp [8][TCH][36];     // wave-private h*C parts

    int blk = blockIdx.x;             // br*64 + b*8 + dgroup
    int br = blk >> 6;
    int b  = (blk >> 3) & 7;
    int d0 = (blk & 7) * 16;          // this block's 16 channels
    int t = threadIdx.x;
    int w = t >> 5, lane = t & 31;
    int s  = lane & 15;               // state index
    int dh = lane >> 4;               // which of the wave's 2 channels
    int dl = 2*w + dh;                // local channel (0..15)
    int d  = d0 + dl;

    const float* dtp = ws + OFF_DT;
    const float* xcp = ws + OFF_XC;
    const float* prp = ws + OFF_PROJ;
    const float* xzp = ws + OFF_XZ;
    float* yzp = ws + OFF_YZ;

    size_t tok0 = ((size_t)br*NB + b) * LSEQ;
    float Arow = -expf(A_log[d*DS + s]);
    float dval = Dp[d];

    // stage one 32-step chunk: 4 arrays x 32 rows x 16 floats = 512 b128 ops
    auto issue = [&](int c, int bufi) {
        size_t tokc = tok0 + (size_t)c*TCH;
        #pragma unroll
        for (int r = 0; r < 2; ++r) {                   // 2 ops per thread
            int i = t + r*256;                          // 0..511
            int a = i >> 7;                             // wave-uniform
            int j = i & 127;
            int l = j >> 2;
            int q = (j & 3) * 4;
            const float* src; float* dst;
            if (a == 0)      { src = dtp + (tokc+l)*DI  + d0 + q;       dst = &sdt[bufi][l][q]; }
            else if (a == 1) { src = xcp + (tokc+l)*DI  + d0 + q;       dst = &sxc[bufi][l][q]; }
            else if (a == 2) { src = prp + (tokc+l)*NXP + DTR + q;      dst = &sB[bufi][l][q]; }
            else             { src = prp + (tokc+l)*NXP + DTR+DS + q;   dst = &sC[bufi][l][q]; }
            cp_b128(dst, src);
        }
    };

    issue(0, 0);
    float h = 0.f;
    for (int c = 0; c < NCH; ++c) {
        int buf = c & 1;
#if USE_ASYNC_LDS
        if (c + 1 < NCH) { issue(c+1, buf^1);           // overlap next fill
                           __builtin_amdgcn_s_wait_asynccnt(2); } // prev chunk landed
        else             { __builtin_amdgcn_s_wait_asynccnt(0); }
#else
        if (c + 1 < NCH) issue(c+1, buf^1);
#endif
        __syncthreads();

        // ---- serial recurrence: only exp+fma on the dependence chain ----
        #pragma unroll 4
        for (int l = 0; l < TCH; ++l) {
            float dtv = sdt[buf][l][dl];                // half-wave broadcast
            float xcv = sxc[buf][l][dl];
            float Bm  = sB[buf][l][s];
            float dA  = expf(dtv * Arow);
            h = fmaf(dA, h, dtv * xcv * Bm);
            sp[w][l][lane] = h * sC[buf][l][s];         // park h*C, no shuffle
        }

        // ---- bulk state reduction + gating (off the serial path) ----
        size_t tokc = tok0 + (size_t)c*TCH;
        #pragma unroll
        for (int p = 0; p < 2; ++p) {
            int lr = (lane & 15) + p*16;                // step this lane reduces
            const float4* q4 = (const float4*)&sp[w][lr][dh*16];
            float4 q0 = q4[0], q1 = q4[1], q2 = q4[2], q3 = q4[3];
            float sum = ((q0.x+q0.y)+(q0.z+q0.w)) + ((q1.x+q1.y)+(q1.z+q1.w))
                      + ((q2.x+q2.y)+(q2.z+q2.w)) + ((q3.x+q3.y)+(q3.z+q3.w));
            float xcv = sxc[buf][lr][dl];
            float y = sum + dval * xcv;
            float zv = xzp[(tokc+lr)*(2*DI) + DI + d];
            y *= zv / (1.f + expf(-zv));                // * silu(z)
            yzp[(tokc+lr)*DI + d] = y;
        }
        __syncthreads();                                // buffer free for refill
    }
}

// =================================================================
// 6) out = yz @ W_out^T + skip*xn, scattered to (B, 4C, H, W)
//    LDS transpose for stride-1 stores along spatial dim
// =================================================================
__global__ __launch_bounds__(128) void gemm_out_kernel(
    const float* __restrict__ W_out, const float* __restrict__ skip_scale,
    float* __restrict__ ws, float* __restrict__ out)
{
    __shared__ float aL[16][132];
    __shared__ float dtile[4][16][17];
    int blk = blockIdx.x;
    int br = blk >> 9, mt = blk & 511;
    size_t tokbase = (size_t)br*TOKB + (size_t)mt*16;
    const float* yz = ws + OFF_YZ + tokbase*DI;
    int t = threadIdx.x;
    for (int idx = t; idx < 16*DI; idx += 128)
        aL[idx >> 7][idx & 127] = yz[idx];
    __syncthreads();
    int wave = t >> 5, lane = t & 31;
    int m = lane & 15, kh = lane >> 4, n = lane & 15;
    int n0 = wave * 16;                                 // 4 waves cover N=64
    v8f acc = {};
    #pragma unroll
    for (int k0 = 0; k0 < DI; k0 += 4) {
        int k = k0 + kh*2;
        v2f a, bb;
        a.x = aL[m][k]; a.y = aL[m][k+1];
        bb.x = W_out[(size_t)(n0+n)*DI + k];
        bb.y = W_out[(size_t)(n0+n)*DI + k + 1];
        acc = __builtin_amdgcn_wmma_f32_16x16x4_f32(
            false, a, false, bb, (short)0, acc, false, false);
    }
    float ss = skip_scale[0];
    const float* xn = ws + OFF_XN + tokbase*CM;
    #pragma unroll
    for (int r = 0; r < 8; ++r) {
        int row = r + 8*kh;
        dtile[wave][row][n] = acc[r] + ss * xn[(size_t)row*CM + n0 + n];
    }
    __syncthreads();
    int tokm = mt * 16;                                 // branch-local token
    int b  = tokm >> 10;
    int l0 = tokm & 1023;
    #pragma unroll
    for (int i = 0; i < 8; ++i) {
        int c = kh*8 + i;                               // tile column
        float v = dtile[wave][lane & 15][c];
        int chan = br*CM + n0 + c;
        out[((size_t)b*(4*CM) + chan)*LSEQ + l0 + (lane & 15)] = v;
    }
}

// =================================================================
extern "C" void kernel_launch(void* const* d_in, const int* in_sizes, int n_in,
                              void* d_out, int out_size, void* d_ws, size_t ws_size,
                              hipStream_t stream)
{
    const float* x0  = (const float*)d_in[0];
    const float* x1  = (const float*)d_in[1];
    const float* x2  = (const float*)d_in[2];
    const float* x3  = (const float*)d_in[3];
    const float* g1  = (const float*)d_in[4];
    const float* be1 = (const float*)d_in[5];
    const float* g2  = (const float*)d_in[6];
    const float* be2 = (const float*)d_in[7];
    const float* g3  = (const float*)d_in[8];
    const float* be3 = (const float*)d_in[9];
    const float* g4  = (const float*)d_in[10];
    const float* be4 = (const float*)d_in[11];
    const float* ssc = (const float*)d_in[12];
    const float* W_in    = (const float*)d_in[13];
    const float* W_conv  = (const float*)d_in[14];
    const float* b_conv  = (const float*)d_in[15];
    const float* W_xproj = (const float*)d_in[16];
    const float* W_dt    = (const float*)d_in[17];
    const float* b_dt    = (const float*)d_in[18];
    const float* A_log   = (const float*)d_in[19];
    const float* Dp      = (const float*)d_in[20];
    const float* W_out   = (const float*)d_in[21];
    float* ws  = (float*)d_ws;
    float* out = (float*)d_out;

    ln_kernel<<<BR*NB*16, 256, 0, stream>>>(x0,x1,x2,x3, g1,g2,g3,g4,
                                            be1,be2,be3,be4, ws);
    gemm_in_kernel<<<BR*512, 256, 0, stream>>>(W_in, ws);
    conv_silu_kernel<<<(BR*TOKB*DI)/256, 256, 0, stream>>>(W_conv, b_conv, ws);
    gemm_proj_kernel<<<BR*512, 128, 0, stream>>>(W_xproj, ws);
    dt_kernel<<<BR*512, 256, 0, stream>>>(W_dt, b_dt, ws);
    scan_kernel<<<256, 256, 0, stream>>>(A_log, Dp, ws);
    gemm_out_kernel<<<BR*512, 128, 0, stream>>>(W_out, ssc, ws, out);
}